// LoRA_72198400245861
// MI455X (gfx1250) — compile-verified
//
#include <hip/hip_runtime.h>

typedef __attribute__((ext_vector_type(2))) float v2f;
typedef __attribute__((ext_vector_type(8))) float v8f;

#define D_EMBED 4096
#define RANK    16
#define M_TOTAL (4 * 4096)              // 16384 rows total (batch*seq)
#define KCHUNKS 8
#define KCHUNK  (D_EMBED / KCHUNKS)     // 512

// ---------------------------------------------------------------------------
// Kernel 0: zero the hr accumulator in workspace (d_ws is poisoned to 0xAA).
// ---------------------------------------------------------------------------
__global__ void lora_zero_hr(float* __restrict__ hr, int n) {
    int i = blockIdx.x * blockDim.x + threadIdx.x;
    if (i < n) hr[i] = 0.0f;
}

// ---------------------------------------------------------------------------
// Kernel 1: hr[m, r] = sum_k h[m, k] * mat_B[r, k]      (h @ B^T)
// One wave per (16-row M tile, 512-wide K chunk); partial 16x16 tile is
// accumulated with 128 chained v_wmma_f32_16x16x4_f32 then atomically added.
// ---------------------------------------------------------------------------
__global__ void lora_stage1(const float* __restrict__ h,
                            const float* __restrict__ matB,
                            float* __restrict__ hr) {
    const int lane   = threadIdx.x & 31;
    const int gwave  = blockIdx.x * (blockDim.x >> 5) + (threadIdx.x >> 5);
    const int mtile  = gwave / KCHUNKS;          // 0 .. M_TOTAL/16 - 1
    const int kchunk = gwave % KCHUNKS;          // 0 .. 7
    const int m0     = mtile * 16;
    const int k0     = kchunk * KCHUNK;

    const int l15  = lane & 15;                  // A: M row / B: N col
    const int koff = (lane >> 4) * 2;            // K pair select (0 or 2)

    // A fragment rows come from h; B fragment: B^T[k][n] = mat_B[n][k]
    const float* hrow = h    + (size_t)(m0 + l15) * D_EMBED;
    const float* brow = matB + (size_t)l15        * D_EMBED;

    v8f acc = {};
#pragma unroll 4
    for (int k = k0; k < k0 + KCHUNK; k += 4) {
        v2f a = *(const v2f*)(hrow + k + koff);
        v2f b = *(const v2f*)(brow + k + koff);
        acc = __builtin_amdgcn_wmma_f32_16x16x4_f32(
            /*neg_a=*/false, a, /*neg_b=*/false, b,
            /*c_mod=*/(short)0, acc, /*reuse_a=*/false, /*reuse_b=*/false);
    }

    // C/D layout: VGPR v, lanes 0-15 -> M = v; lanes 16-31 -> M = v + 8; N = lane&15
    const int mhalf = (lane >> 4) * 8;
#pragma unroll
    for (int v = 0; v < 8; ++v) {
        atomicAdd(&hr[(size_t)(m0 + mhalf + v) * RANK + l15], acc[v]);
    }
}

// ---------------------------------------------------------------------------
// Kernel 2: out[m, d] = sum_r hr[m, r] * mat_A[d, r]    ((h @ B^T) @ A^T)
// One wave per 16x16 output tile; K = RANK = 16 -> 4 WMMA ops.
// hr (1 MB) + mat_A (256 KB) stay resident in L2; stage is store-bound.
// ---------------------------------------------------------------------------
__global__ void lora_stage2(const float* __restrict__ hr,
                            const float* __restrict__ matA,
                            float* __restrict__ out) {
    const int lane   = threadIdx.x & 31;
    const int gwave  = blockIdx.x * (blockDim.x >> 5) + (threadIdx.x >> 5);
    const int ntiles = D_EMBED / 16;             // 256
    const int mtile  = gwave / ntiles;
    const int ntile  = gwave % ntiles;
    const int m0     = mtile * 16;
    const int n0     = ntile * 16;

    const int l15  = lane & 15;
    const int koff = (lane >> 4) * 2;

    // A fragment rows from hr; B fragment: B^T[k][n] = A^T[k][d=n0+n] = mat_A[n0+n][k]
    const float* arow = hr   + (size_t)(m0 + l15) * RANK;
    const float* brow = matA + (size_t)(n0 + l15) * RANK;

    v8f acc = {};
#pragma unroll
    for (int k = 0; k < RANK; k += 4) {
        v2f a = *(const v2f*)(arow + k + koff);
        v2f b = *(const v2f*)(brow + k + koff);
        acc = __builtin_amdgcn_wmma_f32_16x16x4_f32(
            false, a, false, b, (short)0, acc, false, false);
    }

    const int mhalf = (lane >> 4) * 8;
#pragma unroll
    for (int v = 0; v < 8; ++v) {
        out[(size_t)(m0 + mhalf + v) * D_EMBED + n0 + l15] = acc[v];
    }
}

// ---------------------------------------------------------------------------
extern "C" void kernel_launch(void* const* d_in, const int* in_sizes, int n_in,
                              void* d_out, int out_size, void* d_ws, size_t ws_size,
                              hipStream_t stream) {
    const float* h    = (const float*)d_in[0];   // (4, 4096, 4096) f32
    const float* matA = (const float*)d_in[1];   // (4096, 16)       f32
    const float* matB = (const float*)d_in[2];   // (16, 4096)       f32
    float*       out  = (float*)d_out;           // (4, 4096, 4096)  f32
    float*       hr   = (float*)d_ws;            // (16384, 16)      f32  (1 MB)

    // Kernel 0: zero hr accumulator.
    {
        const int n = M_TOTAL * RANK;            // 262144
        lora_zero_hr<<<dim3((n + 255) / 256), dim3(256), 0, stream>>>(hr, n);
    }

    // Kernel 1: hr = h @ B^T (split-K with atomic accumulation).
    {
        const int waves  = (M_TOTAL / 16) * KCHUNKS;   // 1024 * 8 = 8192
        const int blocks = waves / 8;                  // 8 waves (256 thr) / block
        lora_stage1<<<dim3(blocks), dim3(256), 0, stream>>>(h, matB, hr);
    }

    // Kernel 2: out = hr @ A^T.
    {
        const int waves  = (M_TOTAL / 16) * (D_EMBED / 16); // 1024 * 256 = 262144
        const int blocks = waves / 8;                       // 32768
        lora_stage2<<<dim3(blocks), dim3(256), 0, stream>>>(hr, matA, out);
    }
}